// TransD_49727131353817
// MI455X (gfx1250) — compile-verified
//
#include <hip/hip_runtime.h>

typedef __attribute__((ext_vector_type(2))) float v2f;
typedef __attribute__((ext_vector_type(8))) float v8f;

#define ED 128
#define NB 8192
#define TILE 16
#define WPB 4                 // waves per block
#define THREADS (WPB * 32)

__device__ __forceinline__ float renorm_scale(float sq) {
    float n = sqrtf(sq);
    return (n > 1.0f) ? (1.0f / (n + 1e-7f)) : 1.0f;
}

__global__ __launch_bounds__(THREADS)
void transd_kernel(const float* __restrict__ ent,    // entity_emb      [ENT,128]
                   const float* __restrict__ entp,   // entity_emb_p    [ENT,128]
                   const float* __restrict__ rel,    // relation_emb    [REL,128]
                   const float* __restrict__ relp,   // relation_emb_p  [REL,128]
                   const int*  __restrict__ hIdx,
                   const int*  __restrict__ rIdx,
                   const int*  __restrict__ tIdx,
                   float* __restrict__ out)          // [3 * NB * 128] : h_out | rv | t_out
{
    __shared__ float ldsDiag[WPB][2][256];

    const int lane = threadIdx.x & 31;
    const int wave = threadIdx.x >> 5;
    const int tile = blockIdx.x * WPB + wave;
    const int row  = lane & 15;          // sample within tile (M and N index)
    const int lh   = lane >> 4;          // lane half selects K sub-columns
    const int s    = tile * TILE + row;

    const int ih = hIdx[s];
    const int ir = rIdx[s];
    const int it = tIdx[s];

    const float* hp = entp + (long)ih * ED;
    const float* hv = ent  + (long)ih * ED;
    const float* tp = entp + (long)it * ED;
    const float* tv = ent  + (long)it * ED;
    const float* rp = relp + (long)ir * ED;
    const float* rv = rel  + (long)ir * ED;

    // ---- Phase A: gram diagonals via FP32 WMMA + norms via VALU ----
    v8f accH = {};
    v8f accT = {};
    float sq_hp = 0.f, sq_hv = 0.f, sq_tp = 0.f, sq_tv = 0.f;

    #pragma unroll 4
    for (int k0 = 0; k0 < ED; k0 += 4) {
        const int c = k0 + lh * 2;       // this lane's 2 K-columns in [k0,k0+4)
        v2f a  = *(const v2f*)(hp + c);  // A: HP[row][c..c+1]
        v2f b  = *(const v2f*)(hv + c);  // B: HV[row][c..c+1]  (B = HV^T chunk)
        v2f a2 = *(const v2f*)(tp + c);
        v2f b2 = *(const v2f*)(tv + c);
        sq_hp += a.x * a.x + a.y * a.y;
        sq_hv += b.x * b.x + b.y * b.y;
        sq_tp += a2.x * a2.x + a2.y * a2.y;
        sq_tv += b2.x * b2.x + b2.y * b2.y;
        // D = A x B + C, full fp32 MACs, RNE
        accH = __builtin_amdgcn_wmma_f32_16x16x4_f32(false, a,  false, b,  (short)0, accH, false, false);
        accT = __builtin_amdgcn_wmma_f32_16x16x4_f32(false, a2, false, b2, (short)0, accT, false, false);
    }

    // relation-row norms (each lane covers a contiguous 64-float half of its row)
    float sq_rp = 0.f, sq_rv = 0.f;
    {
        const float* rpl = rp + lh * 64;
        const float* rvl = rv + lh * 64;
        #pragma unroll
        for (int i = 0; i < 64; i += 4) {
            float4 x = *(const float4*)(rpl + i);
            float4 y = *(const float4*)(rvl + i);
            sq_rp += x.x * x.x + x.y * x.y + x.z * x.z + x.w * x.w;
            sq_rv += y.x * y.x + y.y * y.y + y.z * y.z + y.w * y.w;
        }
    }

    // combine the two lane-halves of each row's partial sums
    sq_hp += __shfl_xor(sq_hp, 16, 32);
    sq_hv += __shfl_xor(sq_hv, 16, 32);
    sq_tp += __shfl_xor(sq_tp, 16, 32);
    sq_tv += __shfl_xor(sq_tv, 16, 32);
    sq_rp += __shfl_xor(sq_rp, 16, 32);
    sq_rv += __shfl_xor(sq_rv, 16, 32);

    // extract gram diagonals: C/D layout -> VGPR j holds M=j+8*half, N=lane&15
    #pragma unroll
    for (int j = 0; j < 8; ++j) {
        const int m = j + lh * 8;
        ldsDiag[wave][0][m * 16 + row] = accH[j];
        ldsDiag[wave][1][m * 16 + row] = accT[j];
    }
    __syncthreads();
    const float dh = ldsDiag[wave][0][row * 16 + row];  // <hp_row, hv_row>
    const float dt = ldsDiag[wave][1][row * 16 + row];  // <tp_row, tv_row>

    const float s_hp = renorm_scale(sq_hp);
    const float s_hv = renorm_scale(sq_hv);
    const float s_tp = renorm_scale(sq_tp);
    const float s_tv = renorm_scale(sq_tv);
    const float s_rp = renorm_scale(sq_rp);
    const float s_rv = renorm_scale(sq_rv);

    const float ch = s_rp * s_hp * s_hv * dh;  // coefficient on raw rp in h_out
    const float ct = s_rp * s_tp * s_tv * dt;  // coefficient on raw rp in t_out

    // ---- Phase B: coalesced recombination + store ----
    float* out_h  = out;
    float* out_rv = out + (long)NB * ED;
    float* out_t  = out + 2L * (long)NB * ED;

    for (int rw = 0; rw < TILE; ++rw) {
        const int   ihr = __shfl(ih,   rw, 32);
        const int   irr = __shfl(ir,   rw, 32);
        const int   itr = __shfl(it,   rw, 32);
        const float chr = __shfl(ch,   rw, 32);
        const float ctr = __shfl(ct,   rw, 32);
        const float shv = __shfl(s_hv, rw, 32);
        const float stv = __shfl(s_tv, rw, 32);
        const float srv = __shfl(s_rv, rw, 32);

        const float* rpr = relp + (long)irr * ED;
        const float* rvr = rel  + (long)irr * ED;
        const float* hvr = ent  + (long)ihr * ED;
        const float* tvr = ent  + (long)itr * ED;
        const long   ob  = (long)(tile * TILE + rw) * ED;

        #pragma unroll
        for (int q = 0; q < 4; ++q) {
            const int col = lane + q * 32;   // 128B-coalesced per instruction
            const float rpv = rpr[col];
            out_h [ob + col] = chr * rpv + shv * hvr[col];
            out_rv[ob + col] = srv * rvr[col];
            out_t [ob + col] = ctr * rpv + stv * tvr[col];
        }
    }
}

extern "C" void kernel_launch(void* const* d_in, const int* in_sizes, int n_in,
                              void* d_out, int out_size, void* d_ws, size_t ws_size,
                              hipStream_t stream) {
    const float* ent  = (const float*)d_in[0];  // entity_emb
    const float* entp = (const float*)d_in[1];  // entity_emb_p
    const float* rel  = (const float*)d_in[2];  // relation_emb
    const float* relp = (const float*)d_in[3];  // relation_emb_p
    const int*   h    = (const int*)d_in[4];
    const int*   r    = (const int*)d_in[5];
    const int*   t    = (const int*)d_in[6];
    float*       out  = (float*)d_out;

    const int tiles  = NB / TILE;          // 512
    const int blocks = tiles / WPB;        // 128
    transd_kernel<<<blocks, THREADS, 0, stream>>>(ent, entp, rel, relp, h, r, t, out);
}